// Polyblur_34359738723
// MI455X (gfx1250) — compile-verified
//
#include <hip/hip_runtime.h>
#include <math.h>

typedef __attribute__((ext_vector_type(2))) float v2f;
typedef __attribute__((ext_vector_type(8))) float v8f;

#define PB_PI 3.14159265358979323846f
#define PB_SHW 1048   // LDS row stride (max of 8*N1 over both transform sizes)

// ---------------------------------------------------------------------------
// DFT matrix generation: F[k][n] = exp(sign * 2*pi*i * k*n / N1), zero-padded
// to Kpad x Kpad (row-major).
// ---------------------------------------------------------------------------
__global__ void pb_gen_dft(float* __restrict__ Fre, float* __restrict__ Fim,
                           int N1, int Kpad, float sign)
{
    int idx = blockIdx.x * blockDim.x + threadIdx.x;
    if (idx >= Kpad * Kpad) return;
    int k = idx / Kpad;
    int n = idx - k * Kpad;
    float re = 0.f, im = 0.f;
    if (k < N1 && n < N1) {
        int m = (int)(((long long)k * (long long)n) % (long long)N1);
        float ang = sign * 2.0f * PB_PI * (float)m / (float)N1;
        float s, c;
        sincosf(ang, &s, &c);
        re = c; im = s;
    }
    Fre[idx] = re;
    Fim[idx] = im;
}

// ---------------------------------------------------------------------------
// Edge-pad 1024x1024 -> 1048x1048 (ks=12), imaginary part zeroed.
// ---------------------------------------------------------------------------
__global__ void pb_pad_edge(const float* __restrict__ src,
                            float* __restrict__ dstRe, float* __restrict__ dstIm,
                            int Q, int P, int ks)
{
    int idx = blockIdx.x * blockDim.x + threadIdx.x;
    if (idx >= P * P) return;
    int r = idx / P, c = idx - r * P;
    int sr = r - ks; sr = sr < 0 ? 0 : (sr > Q - 1 ? Q - 1 : sr);
    int sc = c - ks; sc = sc < 0 ? 0 : (sc > Q - 1 ? Q - 1 : sc);
    dstRe[idx] = src[sr * Q + sc];
    dstIm[idx] = 0.f;
}

// ---------------------------------------------------------------------------
// p2o padding: place 25x25 psf at origin of PxP, roll by (-12,-12).
// ---------------------------------------------------------------------------
__global__ void pb_psf_pad(const float* __restrict__ psf,
                           float* __restrict__ dstRe, float* __restrict__ dstIm,
                           int P, int kh)
{
    int idx = blockIdx.x * blockDim.x + threadIdx.x;
    if (idx >= P * P) return;
    int r = idx / P, c = idx - r * P;
    int sh = kh / 2;
    int sr = (r + sh) % P;
    int sc = (c + sh) % P;
    float v = 0.f;
    if (sr < kh && sc < kh) v = psf[sr * kh + sc];
    dstRe[idx] = v;
    dstIm[idx] = 0.f;
}

// ---------------------------------------------------------------------------
// Async global->LDS staging helpers (CDNA5 ASYNCcnt path).
// ---------------------------------------------------------------------------
__device__ __forceinline__ void pb_async_g2l_b32(unsigned ldsByteOff, const float* gptr)
{
    asm volatile("global_load_async_to_lds_b32 %0, %1, off"
                 :: "v"(ldsByteOff), "v"(gptr) : "memory");
}
__device__ __forceinline__ void pb_wait_async0()
{
    asm volatile("s_wait_asynccnt 0x0" ::: "memory");
}

// ---------------------------------------------------------------------------
// Mixed-radix stage 1: T[k1][n2 + 8*r] = sum_n1 F[k1][n1] * x_r[8*n1 + n2]
// x_r[j] = in[r*strideB + j*strideT].  Complex, f32 WMMA 16x16x4.
//
// One block (4 waves) per 16-column strip of B.  The strip's B data is exactly
// two full batch-rows of the input (8*N1 == P); they are async-copied to LDS
// once and all M-tiles are computed from LDS by the 4 waves.
// grid = (8*P/16), block = 128.
// ---------------------------------------------------------------------------
__global__ void pb_dft_matmul(const float* __restrict__ inRe, const float* __restrict__ inIm,
                              const float* __restrict__ Fre, const float* __restrict__ Fim,
                              float* __restrict__ Tre, float* __restrict__ Tim,
                              int P, int N1, int Kpad, int strideT, int strideB)
{
    __shared__ float sh[4 * PB_SHW];   // [re r0 | re r1 | im r0 | im r1]

    const int tid   = (int)threadIdx.x;     // 0..127
    const int lane  = tid & 31;
    const int wave  = tid >> 5;             // 0..3
    const int nWav  = (int)blockDim.x >> 5; // 4
    const int nTile = (int)blockIdx.x;
    const int Nw    = 8 * P;
    const int Wst   = 8 * N1;               // == P for both transform sizes
    const int r0    = nTile * 2;            // first of the two batch rows

    // ---- cooperative async staging: 2 rows x {re,im} ----
    {
        const unsigned ldsBase = (unsigned)(size_t)(&sh[0]);
        const float* g0r = inRe + (long long)r0 * strideB;
        const float* g1r = inRe + (long long)(r0 + 1) * strideB;
        const float* g0i = inIm + (long long)r0 * strideB;
        const float* g1i = inIm + (long long)(r0 + 1) * strideB;
        for (int j = tid; j < Wst; j += (int)blockDim.x) {
            const long long go = (long long)j * strideT;
            pb_async_g2l_b32(ldsBase + (unsigned)j * 4u,                  g0r + go);
            pb_async_g2l_b32(ldsBase + (unsigned)(PB_SHW + j) * 4u,       g1r + go);
            pb_async_g2l_b32(ldsBase + (unsigned)(2 * PB_SHW + j) * 4u,   g0i + go);
            pb_async_g2l_b32(ldsBase + (unsigned)(3 * PB_SHW + j) * 4u,   g1i + go);
        }
        pb_wait_async0();
    }
    __syncthreads();

    // keep the tiny, reused F matrix warm
    __builtin_prefetch(Fre, 0, 3);
    __builtin_prefetch(Fim, 0, 3);

    const int nlo  = lane & 15;
    const int koff = (lane >> 4) * 2;       // lanes 0-15: K+0/K+1, lanes 16-31: K+2/K+3
    const int n2   = nlo & 7;               // column -> (n2, dr)
    const int dr   = nlo >> 3;
    const float* shRe = &sh[dr * PB_SHW];
    const float* shIm = &sh[2 * PB_SHW + dr * PB_SHW];

    const int nMT = Kpad / 16;
    for (int mt = wave; mt < nMT; mt += nWav) {
        const int arow = mt * 16 + nlo;     // A-matrix row for this lane

        v8f accRR = {}; // Fr*Xr
        v8f accII = {}; // Fi*Xi
        v8f accI  = {}; // Fr*Xi + Fi*Xr

        for (int k0 = 0; k0 < Kpad; k0 += 4) {
            const int ka = k0 + koff;
            const int kb = ka + 1;

            v2f aR, aI;
            aR.x = Fre[arow * Kpad + ka];
            aR.y = Fre[arow * Kpad + kb];
            aI.x = Fim[arow * Kpad + ka];
            aI.y = Fim[arow * Kpad + kb];

            // branchless zero-padding for n1 >= N1
            const int   c0 = ka < N1 ? ka : (N1 - 1);
            const int   c1 = kb < N1 ? kb : (N1 - 1);
            const float m0 = ka < N1 ? 1.f : 0.f;
            const float m1 = kb < N1 ? 1.f : 0.f;

            v2f bR, bI;
            bR.x = shRe[8 * c0 + n2] * m0;
            bR.y = shRe[8 * c1 + n2] * m1;
            bI.x = shIm[8 * c0 + n2] * m0;
            bI.y = shIm[8 * c1 + n2] * m1;

#if __has_builtin(__builtin_amdgcn_wmma_f32_16x16x4_f32)
            accRR = __builtin_amdgcn_wmma_f32_16x16x4_f32(false, aR, false, bR, (short)0, accRR, false, false);
            accII = __builtin_amdgcn_wmma_f32_16x16x4_f32(false, aI, false, bI, (short)0, accII, false, false);
            accI  = __builtin_amdgcn_wmma_f32_16x16x4_f32(false, aR, false, bI, (short)0, accI,  false, false);
            accI  = __builtin_amdgcn_wmma_f32_16x16x4_f32(false, aI, false, bR, (short)0, accI,  false, false);
#else
            // scalar fallback mirroring the D-matrix layout used at store time
            {
                const int rowBaseF = mt * 16 + ((lane >> 4) * 8);
                for (int kk = 0; kk < 4; ++kk) {
                    int k = k0 + kk;
                    int ck = k < N1 ? k : (N1 - 1);
                    float mk = k < N1 ? 1.f : 0.f;
                    float xr = shRe[8 * ck + n2] * mk;
                    float xi = shIm[8 * ck + n2] * mk;
                    #pragma unroll
                    for (int j = 0; j < 8; ++j) {
                        float fr = Fre[(rowBaseF + j) * Kpad + k];
                        float fi = Fim[(rowBaseF + j) * Kpad + k];
                        accRR[j] += fr * xr;
                        accII[j] += fi * xi;
                        accI[j]  += fr * xi + fi * xr;
                    }
                }
            }
#endif
        }

        // D-matrix layout: VGPR j <-> M = j + 8*(lane>=16), N = lane&15
        const int rowBase = mt * 16 + ((lane >> 4) * 8);
        const int colD    = nTile * 16 + nlo;
        #pragma unroll
        for (int j = 0; j < 8; ++j) {
            int row = rowBase + j;
            if (row < N1) {
                Tre[(long long)row * Nw + colD] = accRR[j] - accII[j];
                Tim[(long long)row * Nw + colD] = accI[j];
            }
        }
    }
}

// ---------------------------------------------------------------------------
// Mixed-radix stage 2: twiddle W_N^{n2*k1} then 8-point DFT over n2.
// out[r*strideB + (k1 + N1*k2)*strideT] = scale * sum_n2 t'[n2] W_8^{n2*k2}
// ---------------------------------------------------------------------------
__global__ void pb_dft_combine(const float* __restrict__ Tre, const float* __restrict__ Tim,
                               float* __restrict__ outRe, float* __restrict__ outIm,
                               int P, int N1, int strideT, int strideB,
                               float sign, float scale)
{
    int idx = blockIdx.x * blockDim.x + threadIdx.x;
    if (idx >= P * N1) return;
    int k1 = idx / P;
    int r  = idx - k1 * P;

    const int Nw = 8 * P;
    const long long base = (long long)k1 * Nw + 8 * r;

    float tr[8], ti[8];
    #pragma unroll
    for (int n2 = 0; n2 < 8; ++n2) {
        float xr = Tre[base + n2];
        float xi = Tim[base + n2];
        float ang = sign * 2.0f * PB_PI * (float)(n2 * k1) / (float)P;
        float s, c;
        sincosf(ang, &s, &c);
        tr[n2] = xr * c - xi * s;
        ti[n2] = xr * s + xi * c;
    }

    float c8[8], s8[8];
    #pragma unroll
    for (int j = 0; j < 8; ++j) {
        float ang = sign * 2.0f * PB_PI * (float)j / 8.0f;
        sincosf(ang, &s8[j], &c8[j]);
    }

    #pragma unroll
    for (int k2 = 0; k2 < 8; ++k2) {
        float sr = 0.f, si = 0.f;
        #pragma unroll
        for (int n2 = 0; n2 < 8; ++n2) {
            int j = (n2 * k2) & 7;
            sr += tr[n2] * c8[j] - ti[n2] * s8[j];
            si += tr[n2] * s8[j] + ti[n2] * c8[j];
        }
        int k = k1 + N1 * k2;
        long long o = (long long)r * strideB + (long long)k * strideT;
        outRe[o] = sr * scale;
        outIm[o] = si * scale;
    }
}

// ---------------------------------------------------------------------------
// Pointwise rank-3 polynomial filter (alpha=2, b=4 -> a3=-1, a2=4, a1=-6).
// ---------------------------------------------------------------------------
__global__ void pb_poly(const float* __restrict__ Yre, const float* __restrict__ Yim,
                        const float* __restrict__ Kre, const float* __restrict__ Kim,
                        float* __restrict__ Xre, float* __restrict__ Xim, int n)
{
    int i = blockIdx.x * blockDim.x + threadIdx.x;
    if (i >= n) return;
    const float a3 = -1.f, a2 = 4.f, a1 = -6.f, bb = 4.f;
    float kr = Kre[i], ki = Kim[i];
    float mag = sqrtf(kr * kr + ki * ki) + 1e-8f;
    float cr = kr / mag, ci = -ki / mag;
    float yr = Yre[i], yi = Yim[i];
    float ycr = cr * yr - ci * yi;
    float yci = cr * yi + ci * yr;
    float xr = a3 * ycr, xi = a3 * yci;
    float t;
    t  = kr * xr - ki * xi + a2 * ycr;
    xi = kr * xi + ki * xr + a2 * yci; xr = t;
    t  = kr * xr - ki * xi + a1 * ycr;
    xi = kr * xi + ki * xr + a1 * yci; xr = t;
    t  = kr * xr - ki * xi + bb * ycr;
    xi = kr * xi + ki * xr + bb * yci; xr = t;
    Xre[i] = xr;
    Xim[i] = xi;
}

__global__ void pb_crop(const float* __restrict__ src, float* __restrict__ dst,
                        int P, int Q, int ks)
{
    int idx = blockIdx.x * blockDim.x + threadIdx.x;
    if (idx >= Q * Q) return;
    int r = idx / Q, c = idx - r * Q;
    dst[idx] = src[(long long)(r + ks) * P + (c + ks)];
}

__global__ void pb_copy_complex(const float* __restrict__ src,
                                float* __restrict__ dstRe, float* __restrict__ dstIm, int n)
{
    int i = blockIdx.x * blockDim.x + threadIdx.x;
    if (i >= n) return;
    dstRe[i] = src[i];
    dstIm[i] = 0.f;
}

__global__ void pb_copy_real(const float* __restrict__ src, float* __restrict__ dst, int n)
{
    int i = blockIdx.x * blockDim.x + threadIdx.x;
    if (i >= n) return;
    dst[i] = src[i];
}

__global__ void pb_grad_mult(const float* __restrict__ Yre, const float* __restrict__ Yim,
                             float* __restrict__ Are, float* __restrict__ Aim,
                             int Q, int axis)
{
    int idx = blockIdx.x * blockDim.x + threadIdx.x;
    if (idx >= Q * Q) return;
    int r = idx / Q, c = idx - r * Q;
    int u = (axis == 0) ? c : r;
    float f = (float)((u < Q / 2) ? u : u - Q) / (float)Q;
    float m = 2.0f * PB_PI * f;
    Are[idx] = -m * Yim[idx];
    Aim[idx] =  m * Yre[idx];
}

__global__ void pb_reduce_nm(const float* __restrict__ gx, const float* __restrict__ gy,
                             float* __restrict__ nM, int n)
{
    __shared__ float sh[256];
    int tid = threadIdx.x;
    int i = blockIdx.x * blockDim.x + tid;
    float v = 0.f;
    if (i < n) v = gx[i] * gx[i] + gy[i] * gy[i];
    sh[tid] = v;
    __syncthreads();
    for (int s = 128; s > 0; s >>= 1) {
        if (tid < s) sh[tid] += sh[tid + s];
        __syncthreads();
    }
    if (tid == 0) atomicAdd(nM, sh[0]);
}

__global__ void pb_final(const float* __restrict__ img, const float* __restrict__ imout,
                         const float* __restrict__ Gxi, const float* __restrict__ Gyi,
                         const float* __restrict__ Gxo, const float* __restrict__ Gyo,
                         const float* __restrict__ nM,
                         float* __restrict__ out, int n)
{
    int i = blockIdx.x * blockDim.x + threadIdx.x;
    if (i >= n) return;
    float M = -Gxi[i] * Gxo[i] - Gyi[i] * Gyo[i];
    float z = M / (nM[0] + M);
    z = fmaxf(z, 0.f);
    float v = z * img[i] + (1.f - z) * imout[i];
    out[i] = fminf(fmaxf(v, 0.f), 1.f);
}

// ---------------------------------------------------------------------------
// Host orchestration
// ---------------------------------------------------------------------------
static inline void pb_fft_pass(hipStream_t s,
                               const float* iR, const float* iI,
                               float* oR, float* oI,
                               float* tR, float* tI,
                               const float* Fre, const float* Fim,
                               int P, int N1, int Kpad, int rowPass,
                               float sign, float scale)
{
    const int Ntot = 8 * P;
    dim3 grid(Ntot / 16), block(128);
    const int sT = rowPass ? 1 : P;
    const int sB = rowPass ? P : 1;
    pb_dft_matmul<<<grid, block, 0, s>>>(iR, iI, Fre, Fim, tR, tI, P, N1, Kpad, sT, sB);
    const int tot = P * N1;
    pb_dft_combine<<<(tot + 255) / 256, 256, 0, s>>>(tR, tI, oR, oI, P, N1, sT, sB, sign, scale);
}

extern "C" void kernel_launch(void* const* d_in, const int* in_sizes, int n_in,
                              void* d_out, int out_size, void* d_ws, size_t ws_size,
                              hipStream_t stream)
{
    (void)in_sizes; (void)n_in; (void)out_size; (void)ws_size;

    const int B = 4, C = 3, Q = 1024, KS = 12, KH = 25;
    const int P = Q + 2 * KS;                 // 1048 = 8 * 131
    const int N1P = 131, KpadP = 144;
    const int N1Q = 128, KpadQ = 128;
    const long long P2 = (long long)P * P;
    const long long Q2 = (long long)Q * Q;

    const float* img_in = (const float*)d_in[0];
    const float* ker_in = (const float*)d_in[1];
    float* out = (float*)d_out;
    float* ws = (float*)d_ws;

    size_t o = 0;
    float* F131f_re = ws + o; o += (size_t)KpadP * KpadP;
    float* F131f_im = ws + o; o += (size_t)KpadP * KpadP;
    float* F131i_re = ws + o; o += (size_t)KpadP * KpadP;
    float* F131i_im = ws + o; o += (size_t)KpadP * KpadP;
    float* F128f_re = ws + o; o += (size_t)KpadQ * KpadQ;
    float* F128f_im = ws + o; o += (size_t)KpadQ * KpadQ;
    float* F128i_re = ws + o; o += (size_t)KpadQ * KpadQ;
    float* F128i_im = ws + o; o += (size_t)KpadQ * KpadQ;
    float* Are = ws + o; o += P2;
    float* Aim = ws + o; o += P2;
    float* Bre = ws + o; o += P2;
    float* Bim = ws + o; o += P2;
    float* Kre = ws + o; o += P2;
    float* Kim = ws + o; o += P2;
    float* Tre = ws + o; o += (size_t)KpadP * 8 * P;
    float* Tim = ws + o; o += (size_t)KpadP * 8 * P;
    float* Yi_re = ws + o; o += Q2;
    float* Yi_im = ws + o; o += Q2;
    float* Yo_re = ws + o; o += Q2;
    float* Yo_im = ws + o; o += Q2;
    float* Gxi = ws + o; o += Q2;
    float* Gyi = ws + o; o += Q2;
    float* Gxo = ws + o; o += Q2;
    float* Gyo = ws + o; o += Q2;
    float* Imout = ws + o; o += Q2;
    float* nM = ws + o; o += 16;

    {
        int tp = KpadP * KpadP, tq = KpadQ * KpadQ;
        pb_gen_dft<<<(tp + 255) / 256, 256, 0, stream>>>(F131f_re, F131f_im, N1P, KpadP, -1.f);
        pb_gen_dft<<<(tp + 255) / 256, 256, 0, stream>>>(F131i_re, F131i_im, N1P, KpadP, +1.f);
        pb_gen_dft<<<(tq + 255) / 256, 256, 0, stream>>>(F128f_re, F128f_im, N1Q, KpadQ, -1.f);
        pb_gen_dft<<<(tq + 255) / 256, 256, 0, stream>>>(F128i_re, F128i_im, N1Q, KpadQ, +1.f);
    }
    hipMemsetAsync(nM, 0, 16 * sizeof(float), stream);

    const int nP2 = (int)P2, nQ2 = (int)Q2;
    const float invP2 = 1.0f / (float)(P2);
    const float invQ2 = 1.0f / (float)(Q2);

    for (int im = 0; im < B * C; ++im) {
        const float* imgP = img_in + (size_t)im * Q2;
        const float* psf  = ker_in + (size_t)im * KH * KH;
        float* outP = out + (size_t)im * Q2;

        // ---- OTF K = fft2(p2o(psf)) ----
        pb_psf_pad<<<(nP2 + 255) / 256, 256, 0, stream>>>(psf, Are, Aim, P, KH);
        pb_fft_pass(stream, Are, Aim, Bre, Bim, Tre, Tim, F131f_re, F131f_im, P, N1P, KpadP, 1, -1.f, 1.f);
        pb_fft_pass(stream, Bre, Bim, Kre, Kim, Tre, Tim, F131f_re, F131f_im, P, N1P, KpadP, 0, -1.f, 1.f);

        // ---- Y = fft2(edge-pad(img)) ----
        pb_pad_edge<<<(nP2 + 255) / 256, 256, 0, stream>>>(imgP, Are, Aim, Q, P, KS);
        pb_fft_pass(stream, Are, Aim, Bre, Bim, Tre, Tim, F131f_re, F131f_im, P, N1P, KpadP, 1, -1.f, 1.f);
        pb_fft_pass(stream, Bre, Bim, Are, Aim, Tre, Tim, F131f_re, F131f_im, P, N1P, KpadP, 0, -1.f, 1.f);

        // ---- X = poly(Y, K, C) ----
        pb_poly<<<(nP2 + 255) / 256, 256, 0, stream>>>(Are, Aim, Kre, Kim, Bre, Bim, nP2);

        // ---- imout = real(ifft2(X)) cropped ----
        pb_fft_pass(stream, Bre, Bim, Are, Aim, Tre, Tim, F131i_re, F131i_im, P, N1P, KpadP, 1, +1.f, 1.f);
        pb_fft_pass(stream, Are, Aim, Bre, Bim, Tre, Tim, F131i_re, F131i_im, P, N1P, KpadP, 0, +1.f, invP2);
        pb_crop<<<(nQ2 + 255) / 256, 256, 0, stream>>>(Bre, Imout, P, Q, KS);

        // ---- halo masking: spectra of img and imout ----
        pb_copy_complex<<<(nQ2 + 255) / 256, 256, 0, stream>>>(imgP, Are, Aim, nQ2);
        pb_fft_pass(stream, Are, Aim, Bre, Bim, Tre, Tim, F128f_re, F128f_im, Q, N1Q, KpadQ, 1, -1.f, 1.f);
        pb_fft_pass(stream, Bre, Bim, Yi_re, Yi_im, Tre, Tim, F128f_re, F128f_im, Q, N1Q, KpadQ, 0, -1.f, 1.f);

        pb_copy_complex<<<(nQ2 + 255) / 256, 256, 0, stream>>>(Imout, Are, Aim, nQ2);
        pb_fft_pass(stream, Are, Aim, Bre, Bim, Tre, Tim, F128f_re, F128f_im, Q, N1Q, KpadQ, 1, -1.f, 1.f);
        pb_fft_pass(stream, Bre, Bim, Yo_re, Yo_im, Tre, Tim, F128f_re, F128f_im, Q, N1Q, KpadQ, 0, -1.f, 1.f);

        // ---- 4 spectral gradients via inverse FFT ----
        const float* Ys_re[4] = { Yi_re, Yi_re, Yo_re, Yo_re };
        const float* Ys_im[4] = { Yi_im, Yi_im, Yo_im, Yo_im };
        const int    axes[4]  = { 0, 1, 0, 1 };
        float*       Gs[4]    = { Gxi, Gyi, Gxo, Gyo };
        for (int g = 0; g < 4; ++g) {
            pb_grad_mult<<<(nQ2 + 255) / 256, 256, 0, stream>>>(Ys_re[g], Ys_im[g], Are, Aim, Q, axes[g]);
            pb_fft_pass(stream, Are, Aim, Bre, Bim, Tre, Tim, F128i_re, F128i_im, Q, N1Q, KpadQ, 1, +1.f, 1.f);
            pb_fft_pass(stream, Bre, Bim, Are, Aim, Tre, Tim, F128i_re, F128i_im, Q, N1Q, KpadQ, 0, +1.f, invQ2);
            pb_copy_real<<<(nQ2 + 255) / 256, 256, 0, stream>>>(Are, Gs[g], nQ2);
        }

        // ---- nM, blend, clip ----
        pb_reduce_nm<<<(nQ2 + 255) / 256, 256, 0, stream>>>(Gxi, Gyi, nM + im, nQ2);
        pb_final<<<(nQ2 + 255) / 256, 256, 0, stream>>>(imgP, Imout, Gxi, Gyi, Gxo, Gyo, nM + im, outP, nQ2);
    }
}